// MedianFilter_26147760898593
// MI455X (gfx1250) — compile-verified
//
#include <hip/hip_runtime.h>
#include <stdint.h>

// Problem constants (reference: xs[B=4,T=32,N=128,D=32], A[1,128,128])
#define B_ 4
#define T_ 32
#define N_ 128
#define D_ 32
#define SLICES 4
#define NODES_PER_BLOCK (N_ / SLICES)              // 32 nodes per block
#define THREADS 256
#define WAVES (THREADS / 32)                       // 8 waves (wave32)
#define NODES_PER_WAVE (NODES_PER_BLOCK / WAVES)   // 4 nodes per wave

// Order-preserving float->uint key: negative floats map below positives,
// monotone w.r.t. float ordering (no NaNs in the input distribution).
__device__ __forceinline__ uint32_t f2key(uint32_t u) {
  return u ^ ((u & 0x80000000u) ? 0xFFFFFFFFu : 0x80000000u);
}

__global__ __launch_bounds__(THREADS)
void median_graph_kernel(const float* __restrict__ xs,
                         const int*   __restrict__ A,
                         float*       __restrict__ out) {
  // LDS: full current frame as sortable keys + per-node compacted neighbor lists
  __shared__ __align__(16) uint32_t keys[N_ * D_];        // 16 KB
  __shared__ uint8_t  nbr[NODES_PER_BLOCK * N_];          // 4 KB
  __shared__ int      deg[NODES_PER_BLOCK];

  const int f     = blockIdx.x;          // flattened (b*T + t)
  const int t     = f & (T_ - 1);
  const int slice = blockIdx.y;
  const int tid   = threadIdx.x;
  const int wave  = tid >> 5;
  const int lane  = tid & 31;            // lane == channel d (D == wave32)

  const float* frame = xs + (size_t)f * (N_ * D_);

  // ---- Stage current frame (16 KB) into LDS via CDNA5 async global->LDS ----
  {
    // Low 32 bits of the flat shared address == LDS byte offset.
    uint32_t lds_base = (uint32_t)(uintptr_t)(&keys[0]);
    for (int c = tid; c < (N_ * D_) / 4; c += THREADS) {   // 1024 x 16B chunks
      uint32_t     dst = lds_base + (uint32_t)c * 16u;
      const float* src = frame + c * 4;
      asm volatile("global_load_async_to_lds_b128 %0, %1, off"
                   :: "v"(dst), "v"(src) : "memory");
    }
    asm volatile("s_wait_asynccnt 0" ::: "memory");
  }

  // ---- Build neighbor lists with wave32 ballot compaction (coalesced A reads) ----
  for (int rr = 0; rr < NODES_PER_WAVE; ++rr) {
    const int ln = wave * NODES_PER_WAVE + rr;           // local node id
    const int n  = slice * NODES_PER_BLOCK + ln;         // global node id
    int base = 0;
    #pragma unroll
    for (int g = 0; g < 4; ++g) {
      const int  j   = g * 32 + lane;
      const bool act = (A[n * N_ + j] != 0) || (j == n); // A_f = A + I
      const uint32_t m = __builtin_amdgcn_ballot_w32(act);
      if (act)
        nbr[ln * N_ + base + __popc(m & ((1u << lane) - 1u))] = (uint8_t)j;
      base += __popc(m);
    }
    if (lane == 0) deg[ln] = base;
  }
  __syncthreads();

  // ---- Convert staged floats to order-preserving keys, in place ----
  for (int i = tid; i < N_ * D_; i += THREADS) keys[i] = f2key(keys[i]);
  __syncthreads();

  const int hp = (t > 0) ? 1 : 0;        // prev-frame self candidate valid
  const int hn = (t < T_ - 1) ? 1 : 0;   // next-frame self candidate valid

  // ---- One wave per node; lane d; lower median via 16-pass radix-4 rank search.
  //      Each pass reads every candidate key ONCE from LDS and compares against
  //      the three 2-bit prefix extensions -> halves DS-issue vs 1-bit search.
  for (int rr = 0; rr < NODES_PER_WAVE; ++rr) {
    const int ln = wave * NODES_PER_WAVE + rr;
    const int n  = slice * NODES_PER_BLOCK + ln;
    const size_t self = (size_t)f * (N_ * D_) + (size_t)n * D_ + lane;

    uint32_t pk = 0u, nk = 0u;
    if (hp) pk = f2key(__float_as_uint(xs[self - (size_t)(N_ * D_)]));
    if (hn) nk = f2key(__float_as_uint(xs[self + (size_t)(N_ * D_)]));

    const int dg = deg[ln];
    const int m  = (dg + hp + hn - 1) >> 1;              // lower-median rank
    const uint32_t* row32 = (const uint32_t*)&nbr[ln * N_];
    const int full = dg >> 2;
    const int rem  = dg & 3;
    const uint8_t* tail = &nbr[ln * N_ + full * 4];

    uint32_t r = 0;
    for (int b = 30; b >= 0; b -= 2) {
      const uint32_t c1 = r | (1u << b);
      const uint32_t c2 = r | (2u << b);
      const uint32_t c3 = r | (3u << b);
      int n1 = (hp & (int)(pk < c1)) + (hn & (int)(nk < c1));
      int n2 = (hp & (int)(pk < c2)) + (hn & (int)(nk < c2));
      int n3 = (hp & (int)(pk < c3)) + (hn & (int)(nk < c3));
      for (int w = 0; w < full; ++w) {                   // 4 packed u8 indices / dword
        const uint32_t q  = row32[w];
        const uint32_t k0 = keys[((q      ) & 255u) * D_ + lane];
        const uint32_t k1 = keys[((q >> 8 ) & 255u) * D_ + lane];
        const uint32_t k2 = keys[((q >> 16) & 255u) * D_ + lane];
        const uint32_t k3 = keys[((q >> 24)       ) * D_ + lane];
        n1 += (int)(k0 < c1) + (int)(k1 < c1) + (int)(k2 < c1) + (int)(k3 < c1);
        n2 += (int)(k0 < c2) + (int)(k1 < c2) + (int)(k2 < c2) + (int)(k3 < c2);
        n3 += (int)(k0 < c3) + (int)(k1 < c3) + (int)(k2 < c3) + (int)(k3 < c3);
      }
      for (int i = 0; i < rem; ++i) {
        const uint32_t kv = keys[(uint32_t)tail[i] * D_ + lane];
        n1 += (int)(kv < c1);
        n2 += (int)(kv < c2);
        n3 += (int)(kv < c3);
      }
      // Largest feasible 2-bit extension (feasible set is downward-closed [0, s_m]).
      if      (n3 <= m) r = c3;
      else if (n2 <= m) r = c2;
      else if (n1 <= m) r = c1;
    }

    const uint32_t ub = (r & 0x80000000u) ? (r ^ 0x80000000u) : ~r;
    out[self] = __uint_as_float(ub);
  }
}

extern "C" void kernel_launch(void* const* d_in, const int* in_sizes, int n_in,
                              void* d_out, int out_size, void* d_ws, size_t ws_size,
                              hipStream_t stream) {
  const float* xs  = (const float*)d_in[0];   // [4,32,128,32] f32
  const int*   A   = (const int*)d_in[1];     // [1,128,128] i32
  float*       out = (float*)d_out;           // [4,32,128,32] f32
  (void)in_sizes; (void)n_in; (void)out_size; (void)d_ws; (void)ws_size;

  dim3 grid(B_ * T_, SLICES);
  median_graph_kernel<<<grid, THREADS, 0, stream>>>(xs, A, out);
}